// Model_39676907886938
// MI455X (gfx1250) — compile-verified
//
#include <hip/hip_runtime.h>
#include <hip/hip_bf16.h>
#include <math.h>

// ---------------------------------------------------------------------------
// Model: conv1x1 (W1) -> LayerNorm(L) -> self-attn (q=k=h1p, v=h1, logits*2000,
// softmax) -> conv1x1 (W2).  fp32 WMMA (V_WMMA_F32_16X16X4_F32) everywhere:
// the *2000 logit scale makes softmax argmax-like, so bf16 would flip results.
//
// Row-major GEMMs: block-cooperative, 8 waves stacked in M share a K x 64
// B-panel staged in LDS by the Tensor Data Mover (tensor_load_to_lds, double
// buffered, TENSORcnt-synchronized).  TDM zero-fills the K tail and applies
// hardware padding (64 dwords + 8 pad = 72-float LDS row stride) so the two
// half-wave K rows hit disjoint LDS bank groups.  Full panels use a fast path
// with immediate-offset A loads (no per-substep address math).
//
// Workspace (floats), required ws_size >= 384,000,000 bytes:
//   h1  : [32,1000,1000]   h1p : [32,1000,1000] (reused for h2)   qk : [32,1000,1000]
// ---------------------------------------------------------------------------

typedef float        v2f __attribute__((ext_vector_type(2)));
typedef float        v8f __attribute__((ext_vector_type(8)));
typedef unsigned int v4u __attribute__((ext_vector_type(4)));
typedef int          v4i __attribute__((ext_vector_type(4)));
typedef int          v8i __attribute__((ext_vector_type(8)));

#define KC          32     // K panel depth per TDM tile
#define LDSS        72     // padded LDS row stride (64 data + 8 pad dwords)
#define TILE_N      64     // block N extent (4 x 16 per wave)
#define BLOCK_M     256    // 8 waves x 32 rows

// ---------------------------------------------------------------------------
// TDM 2D tile load: 64 x KC f32 tile from row-major B (stride ldb elems) into
// LDS at lds_off.  D# built per CDNA5 ISA Sec. 8:
//   group0: count=1 | lds_addr | global_addr(57b) | type=2
//   group1: data_size=4B, pad_enable, pad_interval=64dw, pad_amount=8dw,
//           tensor_dim0/1 = remaining extent (OOB reads return zero),
//           tile_dim0=64, tile_dim1=KC, dim0_stride=ldb
// This toolchain exposes the 6-arg builtin: (g0, g1, g2, g3, g4, cpol).
// ---------------------------------------------------------------------------
__device__ __forceinline__ void tdm_load_panel(unsigned int lds_off,
                                               const float* gptr,
                                               unsigned int rem_n,
                                               unsigned int rem_k,
                                               unsigned int ldb)
{
    const unsigned long long ga = (unsigned long long)(uintptr_t)gptr;

    v4u g0;
    g0[0] = 1u;                                        // count=1 (valid user D#)
    g0[1] = lds_off;                                   // lds_addr (bytes)
    g0[2] = (unsigned int)(ga & 0xffffffffu);          // global_addr[31:0]
    g0[3] = (unsigned int)((ga >> 32) & 0x01ffffffu)   // global_addr[56:32]
          | (2u << 30);                                // type=2 ("image")

    const unsigned int ds_pad = (2u << 16)             // data_size = 4 bytes
                              | (1u << 20)             // pad_enable
                              | (5u << 22)             // pad_interval: 64 dwords
                              | (7u << 25);            // pad_amount: 8 dwords
    v8i g1;
    g1[0] = (int)ds_pad;                               // wg_mask=0 (not in cluster)
    g1[1] = (int)((rem_n & 0xffffu) << 16);            // tensor_dim0[15:0]
    g1[2] = (int)((rem_n >> 16) | ((rem_k & 0xffffu) << 16));   // dim0 hi | dim1 lo
    g1[3] = (int)((rem_k >> 16) | (64u << 16));        // dim1 hi | tile_dim0=64
    g1[4] = (int)(unsigned int)KC;                     // tile_dim1=KC, tile_dim2=0
    g1[5] = (int)ldb;                                  // tensor_dim0_stride[31:0]
    g1[6] = 0;                                         // stride0 hi | stride1 lo
    g1[7] = 0;

    const v4i z4 = {0, 0, 0, 0};                       // groups 2/3 unused (2D)
    const v8i z8 = {0, 0, 0, 0, 0, 0, 0, 0};           // extra group (unused)
    __builtin_amdgcn_tensor_load_to_lds(g0, g1, z4, z4, z8, 0);
}

// ---------------------------------------------------------------------------
// Cooperative WMMA GEMM:  C[b] = alpha * A[b] x B[b] + bias
//   A row-major M x K (lda), B row-major K x N (ldb), C row-major (ldc).
// Block = 256 threads = 8 waves stacked in M; block tile 256 x 64.
// B panel staged by TDM, double buffered; per wave: 2x4 accum tiles,
// 8 v_wmma_f32_16x16x4_f32 per K-substep of 4, B frags read from LDS.
// No early exit: every wave reaches every barrier; EXEC all-ones at WMMA.
// ---------------------------------------------------------------------------
__global__ __launch_bounds__(256) void wmma_gemm_coop(
    const float* __restrict__ A, long long strideA, int lda,
    const float* __restrict__ B, long long strideB, int ldb,
    float* __restrict__ C, long long strideC, int ldc,
    const float* __restrict__ bias, float alpha,
    int M, int N, int K, int mblocks, int nblocks)
{
    __shared__ float Bs[2][KC * LDSS];                 // 18,432 B

    const int tid  = threadIdx.x;
    const int lane = tid & 31;
    const int wave = tid >> 5;

    const int bid = blockIdx.x;
    const int b   = bid / (mblocks * nblocks);
    const int rem = bid % (mblocks * nblocks);
    const int mb  = rem / nblocks;
    const int nb  = rem % nblocks;
    const int mbase = mb * BLOCK_M + wave * 32;
    const int nbase = nb * TILE_N;

    const float* Ab = A + (size_t)b * (size_t)strideA;
    const float* Bb = B + (size_t)b * (size_t)strideB;
    float*       Cb = C + (size_t)b * (size_t)strideC;

    const int khalf = lane >> 4;
    const int ml    = lane & 15;

    const int rowA0 = min(mbase + ml,      M - 1);
    const int rowA1 = min(mbase + 16 + ml, M - 1);
    const size_t aoff0 = (size_t)rowA0 * lda + 2 * khalf;
    const size_t aoff1 = (size_t)rowA1 * lda + 2 * khalf;

    v8f acc[2][4] = {};

    const int nsteps = (K + KC - 1) / KC;
    const unsigned int rem_n = (unsigned int)(N - nbase);

    if (wave == 0)
        tdm_load_panel((unsigned int)(uintptr_t)&Bs[0][0],
                       Bb + nbase, rem_n, (unsigned int)K, (unsigned int)ldb);

    int buf = 0;
    for (int s = 0; s < nsteps; ++s) {
        const int k0 = s * KC;
        if (wave == 0) {
            if (s + 1 < nsteps) {
                tdm_load_panel((unsigned int)(uintptr_t)&Bs[buf ^ 1][0],
                               Bb + (size_t)(k0 + KC) * ldb + nbase,
                               rem_n, (unsigned int)(K - (k0 + KC)),
                               (unsigned int)ldb);
                __builtin_amdgcn_s_wait_tensorcnt(1);  // panel s complete
            } else {
                __builtin_amdgcn_s_wait_tensorcnt(0);
            }
        }
        __syncthreads();                               // panel `buf` visible

        const float* bs = &Bs[buf][0];
        if (k0 + KC <= K) {
            // -------- full panel: immediate-offset A loads, no addr math ----
            const float* pA0 = Ab + aoff0 + k0;
            const float* pA1 = Ab + aoff1 + k0;
#pragma unroll
            for (int ks = 0; ks < KC; ks += 4) {
                const v2f a0 = *(const v2f*)(pA0 + ks);
                const v2f a1 = *(const v2f*)(pA1 + ks);

                const int bb = (ks + 2 * khalf) * LDSS + ml;
                v2f bf[4];
#pragma unroll
                for (int ni = 0; ni < 4; ++ni) {
                    bf[ni].x = bs[bb + ni * 16];
                    bf[ni].y = bs[bb + LDSS + ni * 16];
                }
#pragma unroll
                for (int ni = 0; ni < 4; ++ni) {
                    acc[0][ni] = __builtin_amdgcn_wmma_f32_16x16x4_f32(
                        false, a0, false, bf[ni], (short)0, acc[0][ni], false, false);
                    acc[1][ni] = __builtin_amdgcn_wmma_f32_16x16x4_f32(
                        false, a1, false, bf[ni], (short)0, acc[1][ni], false, false);
                }
            }
        } else {
            // -------- tail panel: clamp A k-offset; B rows are TDM-zeroed ---
#pragma unroll
            for (int ks = 0; ks < KC; ks += 4) {
                const int kA = min(k0 + ks, K - 4);
                const v2f a0 = *(const v2f*)(Ab + aoff0 + kA);
                const v2f a1 = *(const v2f*)(Ab + aoff1 + kA);

                const int bb = (ks + 2 * khalf) * LDSS + ml;
                v2f bf[4];
#pragma unroll
                for (int ni = 0; ni < 4; ++ni) {
                    bf[ni].x = bs[bb + ni * 16];
                    bf[ni].y = bs[bb + LDSS + ni * 16];
                }
#pragma unroll
                for (int ni = 0; ni < 4; ++ni) {
                    acc[0][ni] = __builtin_amdgcn_wmma_f32_16x16x4_f32(
                        false, a0, false, bf[ni], (short)0, acc[0][ni], false, false);
                    acc[1][ni] = __builtin_amdgcn_wmma_f32_16x16x4_f32(
                        false, a1, false, bf[ni], (short)0, acc[1][ni], false, false);
                }
            }
        }
        __syncthreads();                               // done reading `buf`
        buf ^= 1;
    }

    const int half = lane >> 4;
    const int nl   = lane & 15;
#pragma unroll
    for (int mi = 0; mi < 2; ++mi) {
#pragma unroll
        for (int ni = 0; ni < 4; ++ni) {
            const int col = nbase + ni * 16 + nl;
            if (col < N) {
#pragma unroll
                for (int j = 0; j < 8; ++j) {
                    const int row = mbase + mi * 16 + 8 * half + j;
                    if (row < M) {
                        float v = acc[mi][ni][j] * alpha;
                        if (bias) v += bias[row];
                        Cb[(size_t)row * ldc + col] = v;
                    }
                }
            }
        }
    }
}

// ---------------------------------------------------------------------------
// WMMA GEMM, C = alpha * A x X^T (per batch): used for qk = h1p @ h1p^T.
// B-fragment loads are row-gathers of X identical in shape to A loads
// (aligned float2 per lane), L2-resident.  Independent waves, 32x64 tiles.
// ---------------------------------------------------------------------------
__global__ __launch_bounds__(256) void wmma_gemm_xt(
    const float* __restrict__ A, long long strideA, int lda,
    const float* __restrict__ X, long long strideX, int ldx,
    float* __restrict__ C, long long strideC, int ldc,
    float alpha, int M, int N, int K, int batches, int mgroups, int ngroups)
{
    const int lane = threadIdx.x & 31;
    const int wave = threadIdx.x >> 5;
    const long long g = (long long)blockIdx.x * 8 + wave;
    const long long total = (long long)batches * mgroups * ngroups;
    if (g >= total) return;                 // wave-uniform: EXEC stays all-ones

    const int b   = (int)(g / ((long long)mgroups * ngroups));
    const int rem = (int)(g % ((long long)mgroups * ngroups));
    const int mbase = (rem / ngroups) * 32;
    const int nbase = (rem % ngroups) * TILE_N;

    const int khalf = lane >> 4;
    const int ml    = lane & 15;

    const float* Ab = A + (size_t)b * (size_t)strideA;
    const float* Xb = X + (size_t)b * (size_t)strideX;
    float*       Cb = C + (size_t)b * (size_t)strideC;

    const size_t aoff0 = (size_t)min(mbase + ml,      M - 1) * lda + 2 * khalf;
    const size_t aoff1 = (size_t)min(mbase + 16 + ml, M - 1) * lda + 2 * khalf;

    size_t xoff[4];
#pragma unroll
    for (int ni = 0; ni < 4; ++ni)
        xoff[ni] = (size_t)min(nbase + ni * 16 + ml, N - 1) * ldx + 2 * khalf;

    v8f acc[2][4] = {};

    for (int k = 0; k < K; k += 4) {
        const v2f a0 = *(const v2f*)(Ab + aoff0 + k);
        const v2f a1 = *(const v2f*)(Ab + aoff1 + k);
        v2f bf[4];
#pragma unroll
        for (int ni = 0; ni < 4; ++ni)
            bf[ni] = *(const v2f*)(Xb + xoff[ni] + k);
#pragma unroll
        for (int ni = 0; ni < 4; ++ni) {
            acc[0][ni] = __builtin_amdgcn_wmma_f32_16x16x4_f32(
                false, a0, false, bf[ni], (short)0, acc[0][ni], false, false);
            acc[1][ni] = __builtin_amdgcn_wmma_f32_16x16x4_f32(
                false, a1, false, bf[ni], (short)0, acc[1][ni], false, false);
        }
    }

    const int half = lane >> 4;
    const int nl   = lane & 15;
#pragma unroll
    for (int mi = 0; mi < 2; ++mi) {
#pragma unroll
        for (int ni = 0; ni < 4; ++ni) {
            const int col = nbase + ni * 16 + nl;
            if (col < N) {
#pragma unroll
                for (int j = 0; j < 8; ++j) {
                    const int row = mbase + mi * 16 + 8 * half + j;
                    if (row < M)
                        Cb[(size_t)row * ldc + col] = acc[mi][ni][j] * alpha;
                }
            }
        }
    }
}

// ---------------------------------------------------------------------------
// LayerNorm over last dim (gamma/beta indexed by l).  256 threads per row.
// ---------------------------------------------------------------------------
__global__ __launch_bounds__(256) void layernorm_rows(
    const float* __restrict__ h1, float* __restrict__ h1p,
    const float* __restrict__ gamma, const float* __restrict__ beta, int Lr)
{
    __shared__ float s1[256];
    __shared__ float s2[256];
    const int tid = threadIdx.x;
    const float* in = h1 + (size_t)blockIdx.x * Lr;

    float sum = 0.0f, sq = 0.0f;
    for (int i = tid; i < Lr; i += 256) {
        const float v = in[i];
        sum += v;
        sq  += v * v;
    }
    s1[tid] = sum;
    s2[tid] = sq;
    __syncthreads();
    for (int off = 128; off > 0; off >>= 1) {
        if (tid < off) {
            s1[tid] += s1[tid + off];
            s2[tid] += s2[tid + off];
        }
        __syncthreads();
    }
    const float invL = 1.0f / (float)Lr;
    const float mu   = s1[0] * invL;
    const float var  = s2[0] * invL - mu * mu;
    const float rs   = rsqrtf(var + 1e-5f);

    float* out = h1p + (size_t)blockIdx.x * Lr;
    for (int i = tid; i < Lr; i += 256)
        out[i] = (in[i] - mu) * rs * gamma[i] + beta[i];
}

// ---------------------------------------------------------------------------
// In-place row softmax. 256 threads per row.
// ---------------------------------------------------------------------------
__global__ __launch_bounds__(256) void softmax_rows(float* __restrict__ qk, int Lr)
{
    __shared__ float red[256];
    const int tid = threadIdx.x;
    float* p = qk + (size_t)blockIdx.x * Lr;

    float mx = -INFINITY;
    for (int i = tid; i < Lr; i += 256) mx = fmaxf(mx, p[i]);
    red[tid] = mx;
    __syncthreads();
    for (int off = 128; off > 0; off >>= 1) {
        if (tid < off) red[tid] = fmaxf(red[tid], red[tid + off]);
        __syncthreads();
    }
    const float rowmax = red[0];
    __syncthreads();

    float s = 0.0f;
    for (int i = tid; i < Lr; i += 256) {
        const float e = __expf(p[i] - rowmax);
        p[i] = e;
        s += e;
    }
    red[tid] = s;
    __syncthreads();
    for (int off = 128; off > 0; off >>= 1) {
        if (tid < off) red[tid] += red[tid + off];
        __syncthreads();
    }
    const float inv = 1.0f / red[0];
    for (int i = tid; i < Lr; i += 256) p[i] *= inv;
}

// ---------------------------------------------------------------------------
// Host side
// ---------------------------------------------------------------------------
extern "C" void kernel_launch(void* const* d_in, const int* in_sizes, int n_in,
                              void* d_out, int out_size, void* d_ws, size_t ws_size,
                              hipStream_t stream)
{
    (void)in_sizes; (void)n_in; (void)out_size; (void)ws_size;

    const int Bn = 32, Cin = 256, Cout = 256, L = 1000, H = 1000;
    const float logit_scale = 2000.0f;           // 1 / SCALE, SCALE = 1000^-1 / 2

    const float* x     = (const float*)d_in[0];  // [B, Cin, L]
    const float* W1    = (const float*)d_in[1];  // [H, Cin]
    const float* b1    = (const float*)d_in[2];  // [H]
    const float* gamma = (const float*)d_in[3];  // [L]
    const float* beta  = (const float*)d_in[4];  // [L]
    const float* W2    = (const float*)d_in[5];  // [Cout, H]
    const float* b2    = (const float*)d_in[6];  // [Cout]
    float* out = (float*)d_out;                  // [B, Cout, L]

    const size_t BHL = (size_t)Bn * H * L;
    float* h1  = (float*)d_ws;
    float* h1p = h1 + BHL;
    float* qk  = h1p + BHL;
    float* h2  = h1p;                            // reuse: h1p dead after qk

    const int nblocks = (L + TILE_N - 1) / TILE_N;       // 16

    // 1) h1 = W1 @ x[b] + b1          (M=H=1000, N=L, K=Cin)
    {
        const int mblocks = (H + BLOCK_M - 1) / BLOCK_M; // 4
        wmma_gemm_coop<<<Bn * mblocks * nblocks, 256, 0, stream>>>(
            W1, 0LL, Cin,
            x, (long long)Cin * L, L,
            h1, (long long)H * L, L,
            b1, 1.0f, H, L, Cin, mblocks, nblocks);
    }

    // 2) LayerNorm over L
    layernorm_rows<<<Bn * H, 256, 0, stream>>>(h1, h1p, gamma, beta, L);

    // 3) qk = (h1p @ h1p^T) * 2000    (M=N=K=1000)
    {
        int mg = (H + 31) / 32, ng = (H + TILE_N - 1) / TILE_N;
        long long waves = (long long)Bn * mg * ng;
        wmma_gemm_xt<<<(int)((waves + 7) / 8), 256, 0, stream>>>(
            h1p, (long long)H * L, L,
            h1p, (long long)H * L, L,
            qk, (long long)H * H, H,
            logit_scale, H, H, L, Bn, mg, ng);
    }

    // 4) softmax over last dim, in place
    softmax_rows<<<Bn * H, 256, 0, stream>>>(qk, H);

    // 5) h2 = attn @ h1               (M=1000, N=L, K=1000)
    {
        const int mblocks = (H + BLOCK_M - 1) / BLOCK_M; // 4
        wmma_gemm_coop<<<Bn * mblocks * nblocks, 256, 0, stream>>>(
            qk, (long long)H * H, H,
            h1, (long long)H * L, L,
            h2, (long long)H * L, L,
            nullptr, 1.0f, H, L, H, mblocks, nblocks);
    }

    // 6) out = W2 @ h2 + b2           (M=Cout=256, N=L, K=H)
    {
        const int mblocks = (Cout + BLOCK_M - 1) / BLOCK_M; // 1
        wmma_gemm_coop<<<Bn * mblocks * nblocks, 256, 0, stream>>>(
            W2, 0LL, H,
            h2, (long long)H * L, L,
            out, (long long)Cout * L, L,
            b2, 1.0f, Cout, L, H, mblocks, nblocks);
    }
}